// phys_model_65661460021562
// MI455X (gfx1250) — compile-verified
//
#include <hip/hip_runtime.h>
#include <hip/hip_bf16.h>
#include <stdint.h>
#include <math.h>

// ---------------- problem constants (from reference) ----------------
#define NBATCH  1024
#define NJ      25          // B
#define NBODY   26          // B+1
#define SPF_K   10
#define NFRAME  15          // 150/10
#define NSNAP   16
#define TAUW    31          // 6+B floats per torque/ref row
#define RFW     156         // 26*6 floats per res_f row
#define DT_     0.0005f
#define POSW    182         // 26*7
#define VELW    156         // 26*6
#define VELBASE ((size_t)NSNAP * NBATCH * POSW)

// ---------------- K1 LDS layout (dwords, per wave) ----------------
#define SUBSTRIDE 224                    // 32 (tau) + 32 (ref) + 160 (rf), padded
#define BUFDW    (SPF_K * SUBSTRIDE)     // 2240 per frame buffer
#define NBUF     3                       // triple buffer -> 2 frames in flight
#define WAVEDW   (NBUF*BUFDW + 4)        // 6724, multiple of 4 (16B-aligned stride)
#define WPB      2                       // waves per block
#define TPB      (WPB*32)

// ---------------- CDNA5 async global->LDS path ----------------
#if __has_builtin(__builtin_amdgcn_global_load_async_to_lds_b32) && \
    __has_builtin(__builtin_amdgcn_global_load_async_to_lds_b128)
#define HAVE_ASYNC 1
#else
#define HAVE_ASYNC 0
#warning "gfx1250 async global->LDS builtins not found; using synchronous LDS staging fallback"
#endif

typedef int v4i __attribute__((vector_size(16)));
typedef int   __attribute__((address_space(1))) gas_i32;
typedef int   __attribute__((address_space(3))) las_i32;
typedef v4i   __attribute__((address_space(1))) gas_v4i;
typedef v4i   __attribute__((address_space(3))) las_v4i;

__device__ __forceinline__ void cp_b32(const float* g, float* l) {
#if HAVE_ASYNC
  __builtin_amdgcn_global_load_async_to_lds_b32(
      (gas_i32*)(uintptr_t)g, (las_i32*)(uint32_t)(uintptr_t)l, 0, 0);
#else
  *l = *g;
#endif
}
__device__ __forceinline__ void cp_b128(const float* g, float* l) {
#if HAVE_ASYNC
  __builtin_amdgcn_global_load_async_to_lds_b128(
      (gas_v4i*)(uintptr_t)g, (las_v4i*)(uint32_t)(uintptr_t)l, 0, 0);
#else
  *(float4*)l = *(const float4*)g;
#endif
}

#define WAIT_ASYNC(n) asm volatile("s_wait_asynccnt %0" :: "i"(n) : "memory")

// ============================================================================
// K1: time integration (wave-per-batch).  Streams torques/refs/res_f through
// LDS with async DMA (2 frames in flight + L2 prefetch 1 frame deeper).
// Emits per-frame snapshots into d_out: body-0 rows in final form, and
// jq/jqd stashed at wp_pos[f][n][1+j][0] / wp_vel[f][n][1+j][0] for K2.
// ============================================================================
__global__ __launch_bounds__(TPB) void phys_dyn_kernel(
    const float* __restrict__ q_init,     // (1024, 32)
    const float* __restrict__ qd_init,    // (1024, 31)
    const float* __restrict__ torques,    // (150, 1024, 31)
    const float* __restrict__ res_f,      // (150, 1024, 26, 6)
    const float* __restrict__ refs,       // (150, 1024, 31)
    const float* __restrict__ tke,        // (1024, 25)
    const float* __restrict__ tkd,        // (1024, 25)
    const float* __restrict__ bmass,      // (1024, 26)
    const float* __restrict__ jaxes,      // (25, 3)
    float* __restrict__ out)
{
  __shared__ __align__(16) float smem[WPB * WAVEDW];

  const int lane  = threadIdx.x & 31;
  const int wslot = threadIdx.x >> 5;
  const int n     = blockIdx.x * WPB + wslot;     // batch id
  if (n >= NBATCH) return;

  float* wbase = smem + wslot * WAVEDW;

  // ---------------- initial state ----------------
  const float* qi  = q_init  + n * 32;
  const float* qdi = qd_init + n * 31;
  float px = qi[0], py = qi[1], pz = qi[2];
  float qx = qi[3], qy = qi[4], qz = qi[5], qw = qi[6];
  {
    float nn = sqrtf(qx*qx + qy*qy + qz*qz + qw*qw);
    float inv = 1.0f / fmaxf(nn, 1e-8f);
    qx *= inv; qy *= inv; qz *= inv; qw *= inv;
  }
  float wx = qdi[0], wy = qdi[1], wz = qdi[2];
  float vx = qdi[3], vy = qdi[4], vz = qdi[5];

  const int b = (lane < NJ) ? lane : (NJ - 1);    // joint owned by this lane
  float jq  = qi[7 + b];
  float jqd = qdi[6 + b];
  const float ke  = tke[n * NJ + b];
  const float kd  = tkd[n * NJ + b];
  const float im0 = 1.0f / (bmass[n * NBODY] + 0.1f);
  const float imb = 1.0f / (bmass[n * NBODY + 1 + b] + 0.1f);
  float ax = jaxes[b*3+0], ay = jaxes[b*3+1], az = jaxes[b*3+2];
  {
    float nn = sqrtf(ax*ax + ay*ay + az*az);
    float inv = 1.0f / fmaxf(nn, 1e-8f);
    ax *= inv; ay *= inv; az *= inv;
  }

  // ---------------- helpers ----------------
  auto issue_frame = [&](int f) {                  // 40 async DMA instructions
    const int s0 = f * SPF_K;
    float* buf = wbase + (f % NBUF) * BUFDW;
    for (int s = 0; s < SPF_K; ++s) {
      const size_t row = (size_t)(s0 + s) * NBATCH + n;
      const float* tg = torques + row * TAUW;
      const float* rg = refs    + row * TAUW;
      const float* fg = res_f   + row * RFW;       // 624B rows: 16B aligned
      float* lb = buf + s * SUBSTRIDE;
      if (lane < TAUW) {
        cp_b32(tg + lane, lb + lane);              // tau row (31 dw, 4B aligned)
        cp_b32(rg + lane, lb + 32 + lane);         // ref row
      }
      cp_b128(fg + lane * 4,       lb + 64 + lane * 4);        // rf dwords 0..127
      if (lane < 7)
        cp_b128(fg + 128 + lane*4, lb + 64 + 128 + lane * 4);  // rf dwords 128..155
    }
  };

  auto prefetch_frame = [&](int f) {               // warm L2 (192MB) one frame deeper
    const int s0 = f * SPF_K;
    for (int s = 0; s < SPF_K; ++s) {
      const size_t row = (size_t)(s0 + s) * NBATCH + n;
      const float* pa;
      if (lane < 2)      pa = torques + row * TAUW + lane * 24;
      else if (lane < 4) pa = refs    + row * TAUW + (lane - 2) * 24;
      else               pa = res_f   + row * RFW  + ((lane - 4) % 5) * 32;
      __builtin_prefetch(pa, 0, 1);
    }
  };

  auto snap_write = [&](int fo) {
    float* gp = out + ((size_t)fo * NBATCH + n) * POSW;
    float* gv = out + VELBASE + ((size_t)fo * NBATCH + n) * VELW;
    if (lane == 0) {                               // body 0 row: final output
      gp[0]=px; gp[1]=py; gp[2]=pz; gp[3]=qx; gp[4]=qy; gp[5]=qz; gp[6]=qw;
      gv[0]=wx; gv[1]=wy; gv[2]=wz; gv[3]=vx; gv[4]=vy; gv[5]=vz;
    }
    if (lane < NJ) {                               // stash jq/jqd for K2
      gp[(1 + lane) * 7] = jq;
      gv[(1 + lane) * 6] = jqd;
    }
  };

  // ---------------- pipeline ----------------
  issue_frame(0);
  snap_write(0);                 // snapshot 0 overlaps frame-0 DMA
  issue_frame(1);
  prefetch_frame(2);

  for (int f = 0; f < NFRAME; ++f) {
    // in-order ASYNC completion: <=40 outstanding => oldest frame group done
    if (f < NFRAME - 1) { WAIT_ASYNC(40); } else { WAIT_ASYNC(0); }
    if (f + 2 < NFRAME) {
      issue_frame(f + 2);
      if (f + 3 < NFRAME) prefetch_frame(f + 3);
    }

    const float* buf = wbase + (f % NBUF) * BUFDW;
    for (int k = 0; k < SPF_K; ++k) {
      const float* sb = buf + k * SUBSTRIDE;
      const float t0 = sb[0], t1 = sb[1], t2 = sb[2];
      const float t3 = sb[3], t4 = sb[4], t5 = sb[5];
      const float r0 = sb[64+0], r1 = sb[64+1], r2 = sb[64+2];
      const float r3 = sb[64+3], r4 = sb[64+4], r5 = sb[64+5];
      // root linear / angular velocity
      vx += DT_ * ((t3 + r3) * im0);
      vy += DT_ * ((t4 + r4) * im0);
      vz += DT_ * ((t5 + r5) * im0 - 9.81f);
      wx += DT_ * (t0 + r0) * im0;
      wy += DT_ * (t1 + r1) * im0;
      wz += DT_ * (t2 + r2) * im0;
      // position (uses updated v)
      px += DT_ * vx; py += DT_ * vy; pz += DT_ * vz;
      // quaternion integration: q = normalize(q + 0.5*DT*([w,0] * q))
      const float mx =  wx*qw + wy*qz - wz*qy;
      const float my = -wx*qz + wy*qw + wz*qx;
      const float mz =  wx*qy - wy*qx + wz*qw;
      const float mw = -(wx*qx + wy*qy + wz*qz);
      const float h = 0.5f * DT_;
      float uqx = qx + h*mx, uqy = qy + h*my, uqz = qz + h*mz, uqw = qw + h*mw;
      float nn = sqrtf(uqx*uqx + uqy*uqy + uqz*uqz + uqw*uqw);
      float inv = 1.0f / fmaxf(nn, 1e-8f);
      qx = uqx*inv; qy = uqy*inv; qz = uqz*inv; qw = uqw*inv;
      // joint DOF (lane-parallel, decoupled)
      const float* rfb = sb + 64 + (1 + b) * 6;
      const float tj = sb[6 + b] + ke * (sb[32 + 6 + b] - jq) - kd * jqd
                     + rfb[0]*ax + rfb[1]*ay + rfb[2]*az;
      jqd += DT_ * tj * imb;
      jq  += DT_ * jqd;
    }
    snap_write(f + 1);
  }
}

// ============================================================================
// K2: forward kinematics.  One thread per (snapshot, batch); lane = consecutive
// batch -> full 32/32 lane utilization, one sincos instruction per joint for
// 32 chains.  Reads the stashed jq/jqd (same-thread RAW before overwrite).
// ============================================================================
__global__ __launch_bounds__(256) void phys_fk_kernel(
    const float* __restrict__ jaxes,      // (25, 3)
    const float* __restrict__ joffs,      // (25, 3)
    float* __restrict__ out)
{
  const int tid = blockIdx.x * 256 + threadIdx.x;
  if (tid >= NSNAP * NBATCH) return;
  const int fo = tid >> 10;           // snapshot
  const int n  = tid & (NBATCH - 1);  // batch (consecutive across lanes)

  float* gp = out + ((size_t)fo * NBATCH + n) * POSW;
  float* gv = out + VELBASE + ((size_t)fo * NBATCH + n) * VELW;

  // root state (body 0 row, already final)
  float cpx = gp[0], cpy = gp[1], cpz = gp[2];
  float cqx = gp[3], cqy = gp[4], cqz = gp[5], cqw = gp[6];
  float cwx = gv[0], cwy = gv[1], cwz = gv[2];
  float cvx = gv[3], cvy = gv[4], cvz = gv[5];

  for (int j = 0; j < NJ; ++j) {
    float* sp = gp + (1 + j) * 7;
    float* sv = gv + (1 + j) * 6;
    const float th  = sp[0];          // stashed jq  (read before overwrite)
    const float thd = sv[0];          // stashed jqd
    // joint constants (wave-uniform j -> scalarizable loads)
    float jax = jaxes[j*3+0], jay = jaxes[j*3+1], jaz = jaxes[j*3+2];
    {
      float nn = sqrtf(jax*jax + jay*jay + jaz*jaz);
      float inv = 1.0f / fmaxf(nn, 1e-8f);
      jax *= inv; jay *= inv; jaz *= inv;
    }
    const float jox = joffs[j*3+0], joy = joffs[j*3+1], joz = joffs[j*3+2];
    // off_w = R(cq)*off
    float tx = 2.f*(cqy*joz - cqz*joy);
    float ty = 2.f*(cqz*jox - cqx*joz);
    float tz = 2.f*(cqx*joy - cqy*jox);
    float owx = jox + cqw*tx + (cqy*tz - cqz*ty);
    float owy = joy + cqw*ty + (cqz*tx - cqx*tz);
    float owz = joz + cqw*tz + (cqx*ty - cqy*tx);
    // axis_w = R(cq)*axis
    float ux = 2.f*(cqy*jaz - cqz*jay);
    float uy = 2.f*(cqz*jax - cqx*jaz);
    float uz = 2.f*(cqx*jay - cqy*jax);
    float awx = jax + cqw*ux + (cqy*uz - cqz*uy);
    float awy = jay + cqw*uy + (cqz*ux - cqx*uz);
    float awz = jaz + cqw*uz + (cqx*uy - cqy*ux);
    // dq = (axis*sin(th/2), cos(th/2)); q_c = cq * dq
    float sh, ch; sincosf(0.5f * th, &sh, &ch);
    const float dqx = jax*sh, dqy = jay*sh, dqz = jaz*sh, dqw = ch;
    const float nqx = cqw*dqx + cqx*dqw + cqy*dqz - cqz*dqy;
    const float nqy = cqw*dqy - cqx*dqz + cqy*dqw + cqz*dqx;
    const float nqz = cqw*dqz + cqx*dqy - cqy*dqx + cqz*dqw;
    const float nqw = cqw*dqw - cqx*dqx - cqy*dqy - cqz*dqz;
    // v_c = cv + cw x off_w   (parent w!)
    const float nvx = cvx + (cwy*owz - cwz*owy);
    const float nvy = cvy + (cwz*owx - cwx*owz);
    const float nvz = cvz + (cwx*owy - cwy*owx);
    // advance chain
    cpx += owx; cpy += owy; cpz += owz;
    cqx = nqx; cqy = nqy; cqz = nqz; cqw = nqw;
    cwx += awx * thd; cwy += awy * thd; cwz += awz * thd;
    cvx = nvx; cvy = nvy; cvz = nvz;
    // body (j+1) output row
    sp[0]=cpx; sp[1]=cpy; sp[2]=cpz; sp[3]=cqx; sp[4]=cqy; sp[5]=cqz; sp[6]=cqw;
    sv[0]=cwx; sv[1]=cwy; sv[2]=cwz; sv[3]=cvx; sv[4]=cvy; sv[5]=cvz;
  }
}

extern "C" void kernel_launch(void* const* d_in, const int* in_sizes, int n_in,
                              void* d_out, int out_size, void* d_ws, size_t ws_size,
                              hipStream_t stream) {
  (void)in_sizes; (void)n_in; (void)out_size; (void)d_ws; (void)ws_size;
  phys_dyn_kernel<<<dim3(NBATCH / WPB), dim3(TPB), 0, stream>>>(
      (const float*)d_in[0], (const float*)d_in[1], (const float*)d_in[2],
      (const float*)d_in[3], (const float*)d_in[4], (const float*)d_in[5],
      (const float*)d_in[6], (const float*)d_in[7], (const float*)d_in[8],
      (float*)d_out);
  phys_fk_kernel<<<dim3((NSNAP * NBATCH) / 256), dim3(256), 0, stream>>>(
      (const float*)d_in[8], (const float*)d_in[9], (float*)d_out);
}